// SPPoolMean_91010357002793
// MI455X (gfx1250) — compile-verified
//
#include <hip/hip_runtime.h>
#include <stdint.h>

// Superpixel mean pooling for [16,32,256,256] f32 with int labels in [0,512).
// Row-parallel: one block per (b,c) row of 65536 elements, 512 label bins.
#define NUM_LABELS 512

template <int BS>
__global__ __launch_bounds__(BS) void sppool_mean_kernel(
    const float* __restrict__ src,
    const int* __restrict__ labels,
    float* __restrict__ out,
    int row_len) {
  __shared__ float s_sum[NUM_LABELS];
  __shared__ float s_cnt[NUM_LABELS];
  __shared__ int s_lab[2][BS * 4];  // double-buffered async label staging

  const int tid = threadIdx.x;
  const long long base = (long long)blockIdx.x * (long long)row_len;
  const float4* __restrict__ src4 = reinterpret_cast<const float4*>(src + base);
  const int4* __restrict__ lab4 = reinterpret_cast<const int4*>(labels + base);
  float4* __restrict__ out4 = reinterpret_cast<float4*>(out + base);
  const int n4 = row_len >> 2;  // float4 elements per row

  // ---- Phase 0: zero histogram ----------------------------------------
  for (int i = tid; i < NUM_LABELS; i += BS) {
    s_sum[i] = 0.0f;
    s_cnt[i] = 0.0f;
  }
  __syncthreads();

  // ---- Phase 1: scatter-accumulate sums & counts (LDS atomics) --------
#pragma unroll 2
  for (int i = tid; i < n4; i += BS) {
    // gfx1250 prefetch a few tiles ahead; locality 3 -> WGP scope so the
    // line lands in near caches for the b128 load that follows shortly.
    __builtin_prefetch(&src4[i + 8 * BS], 0, 3);
    __builtin_prefetch(&lab4[i + 8 * BS], 0, 3);
    float4 v = src4[i];
    int4 l = lab4[i];
    atomicAdd(&s_sum[l.x], v.x);
    atomicAdd(&s_cnt[l.x], 1.0f);
    atomicAdd(&s_sum[l.y], v.y);
    atomicAdd(&s_cnt[l.y], 1.0f);
    atomicAdd(&s_sum[l.z], v.z);
    atomicAdd(&s_cnt[l.z], 1.0f);
    atomicAdd(&s_sum[l.w], v.w);
    atomicAdd(&s_cnt[l.w], 1.0f);
  }
  __syncthreads();

  // ---- Phase 2: per-label mean (in place) -----------------------------
  for (int i = tid; i < NUM_LABELS; i += BS) {
    s_sum[i] = s_sum[i] / s_cnt[i];
  }
  __syncthreads();

  // ---- Phase 3: gather means back -------------------------------------
  // Labels re-streamed via double-buffered async global->LDS loads
  // (gfx1250 GLOBAL_LOAD_ASYNC_TO_LDS_B128, tracked by ASYNCcnt).
  // Each lane consumes exactly the 16 bytes it staged, so per-wave
  // ASYNCcnt waits suffice -- no block barrier needed in this loop.
  const int ntiles = n4 / BS;  // 65536/4/512 = 32
  const uint32_t lds0 = (uint32_t)(uintptr_t)&s_lab[0][tid * 4];
  const uint32_t lds1 = (uint32_t)(uintptr_t)&s_lab[1][tid * 4];

  {  // issue tile 0
    uint64_t ga = (uint64_t)(uintptr_t)&lab4[tid];
    asm volatile("global_load_async_to_lds_b128 %0, %1, off" ::"v"(lds0),
                 "v"(ga)
                 : "memory");
  }
  for (int t = 0; t < ntiles; ++t) {
    const uint32_t nxt = (t & 1) ? lds0 : lds1;
    if (t + 1 < ntiles) {
      uint64_t ga = (uint64_t)(uintptr_t)&lab4[(t + 1) * BS + tid];
      asm volatile("global_load_async_to_lds_b128 %0, %1, off" ::"v"(nxt),
                   "v"(ga)
                   : "memory");
      // async-load completion is in-order: <=1 outstanding => tile t landed.
      asm volatile("s_wait_asynccnt 0x1" ::: "memory");
    } else {
      asm volatile("s_wait_asynccnt 0x0" ::: "memory");
    }
    const int* myl = (t & 1) ? &s_lab[1][tid * 4] : &s_lab[0][tid * 4];
    int4 l = *reinterpret_cast<const int4*>(myl);  // ds_load_b128
    float4 m;
    m.x = s_sum[l.x];
    m.y = s_sum[l.y];
    m.z = s_sum[l.z];
    m.w = s_sum[l.w];
    out4[t * BS + tid] = m;  // global_store_b128
  }
}

extern "C" void kernel_launch(void* const* d_in, const int* in_sizes, int n_in,
                              void* d_out, int out_size, void* d_ws,
                              size_t ws_size, hipStream_t stream) {
  const float* src = (const float*)d_in[0];
  const int* labels = (const int*)d_in[1];
  float* out = (float*)d_out;

  const int row_len = 256 * 256;           // H*W
  const int rows = in_sizes[0] / row_len;  // B*C = 512

  sppool_mean_kernel<512><<<rows, 512, 0, stream>>>(src, labels, out, row_len);
}